// UniformAggregationPure_15040975470960
// MI455X (gfx1250) — compile-verified
//
#include <hip/hip_runtime.h>

#define HIDDEN 128

typedef __attribute__((ext_vector_type(2))) float v2f;
typedef __attribute__((ext_vector_type(8))) float v8f;

// ---------------------------------------------------------------------------
// Zero the sum buffer (d_out) and the counts buffer (in d_ws).
// ---------------------------------------------------------------------------
__global__ void ua_init_kernel(float* __restrict__ out, float* __restrict__ counts,
                               int out_n, int cnt_n) {
    int i = blockIdx.x * blockDim.x + threadIdx.x;
    if (i < out_n) out[i] = 0.0f;
    if (i < cnt_n) counts[i] = 0.0f;
}

// ---------------------------------------------------------------------------
// One wave32 per edge; each lane owns 4 consecutive features (float4).
// Per edge:  2 wave-uniform index loads (single-cacheline broadcast),
//            one coalesced 512B gather (global_load_b128 per lane),
//            4 hardware f32 atomic adds per lane (global_atomic_add_f32,
//            non-returning -> STOREcnt, RMW resolved in the 192MB L2).
// Lane q==0 also bumps the hyperedge count.
// ---------------------------------------------------------------------------
__global__ void ua_scatter_kernel(const float* __restrict__ feats,
                                  const int* __restrict__ hidx,
                                  float* __restrict__ sums,
                                  float* __restrict__ counts,
                                  int n_edges) {
    long long i = (long long)blockIdx.x * blockDim.x + threadIdx.x;
    long long total = (long long)n_edges * 32;
    if (i >= total) return;
    int edge = (int)(i >> 5);            // one wave per edge
    int q    = (int)(i & 31);            // lane -> feature quad
    int node = hidx[edge];               // hyperedge_index[0][edge]
    int he   = hidx[n_edges + edge];     // hyperedge_index[1][edge]

    const float4 v = *(const float4*)(feats + (long long)node * HIDDEN + q * 4);
    float* dst = sums + (long long)he * HIDDEN + q * 4;
    unsafeAtomicAdd(dst + 0, v.x);
    unsafeAtomicAdd(dst + 1, v.y);
    unsafeAtomicAdd(dst + 2, v.z);
    unsafeAtomicAdd(dst + 3, v.w);
    if (q == 0) {
        unsafeAtomicAdd(&counts[he], 1.0f);
    }
}

// ---------------------------------------------------------------------------
// Epilogue: out[he, n] = sums[he, n] / max(count[he], 1)
// Implemented as D = diag(r) * S on 16x16 f32 tiles with V_WMMA_F32_16X16X4_F32.
// One wave per tile; K=16 diagonal contraction split into four K=4 WMMAs with
// the f32 accumulator chained.  VGPR layouts per CDNA5 ISA 7.12.2:
//   A (16x4 f32, 2 VGPRs): lanes 0-15 -> M=lane, VGPR j = K j
//                          lanes 16-31 -> M=lane-16, VGPR j = K j+2
//   B (4x16 f32, 2 VGPRs): lanes 0-15 -> N=lane, VGPR j = K j
//                          lanes 16-31 -> N=lane-16, VGPR j = K j+2
//   C/D (16x16 f32, 8 VGPRs): VGPR v = row v (lanes 0-15, N=lane)
//                             VGPR v = row v+8 (lanes 16-31, N=lane-16)
// ---------------------------------------------------------------------------
__global__ void ua_finalize_wmma_kernel(float* __restrict__ out,
                                        const float* __restrict__ counts,
                                        int num_he) {
    const int lane = threadIdx.x & 31;
    const int wave = (int)(((long long)blockIdx.x * blockDim.x + threadIdx.x) >> 5);
    const int col_tiles = HIDDEN / 16;                 // 8
    const int row_tiles = (num_he + 15) / 16;
    if (wave >= row_tiles * col_tiles) return;

    const int he_base   = (wave / col_tiles) * 16;
    const int feat_base = (wave % col_tiles) * 16;
    const int m = lane & 15;                            // row (A/cnt) & col (B/D)

    if (he_base + 16 <= num_he) {
        // Full tile: WMMA diagonal-scale (EXEC is all 1s here).
        float c = counts[he_base + m];
        float r = 1.0f / fmaxf(c, 1.0f);
        const int khi = (lane < 16) ? 0 : 2;            // K offset for this half
        v8f acc = {};
#pragma unroll
        for (int kc = 0; kc < 4; ++kc) {
            // A_kc[m, k] = r  iff  m == kc*4 + k  else 0
            v2f a;
            a.x = (m == kc * 4 + khi + 0) ? r : 0.0f;
            a.y = (m == kc * 4 + khi + 1) ? r : 0.0f;
            // B_kc[k, n] = S[he_base + kc*4 + k, feat_base + n]
            const float* rowp =
                out + (long long)(he_base + kc * 4 + khi) * HIDDEN + feat_base + m;
            v2f b;
            b.x = rowp[0];
            b.y = rowp[HIDDEN];
            acc = __builtin_amdgcn_wmma_f32_16x16x4_f32(
                false, a, false, b, (short)0, acc, false, false);
        }
        const int rowoff = (lane < 16) ? 0 : 8;
#pragma unroll
        for (int v = 0; v < 8; ++v) {
            out[(long long)(he_base + rowoff + v) * HIDDEN + feat_base + m] = acc[v];
        }
    } else {
        // Tail tile (unused for NUM_HE=50000 which is a multiple of 16):
        // scalar exact divide on valid rows.
        for (int rr = 0; rr < 16; ++rr) {
            int row = he_base + rr;
            if (row >= num_he) break;
            if (lane < 16) {
                float c = counts[row];
                long long o = (long long)row * HIDDEN + feat_base + lane;
                out[o] = out[o] / fmaxf(c, 1.0f);
            }
        }
    }
}

extern "C" void kernel_launch(void* const* d_in, const int* in_sizes, int n_in,
                              void* d_out, int out_size, void* d_ws, size_t ws_size,
                              hipStream_t stream) {
    (void)n_in; (void)ws_size;
    const float* feats = (const float*)d_in[0];       // [N_NODES, 128] f32
    const int*   hidx  = (const int*)d_in[1];         // [2, N_EDGES] int
    float* out    = (float*)d_out;                    // [NUM_HE, 128] f32
    float* counts = (float*)d_ws;                     // [NUM_HE] f32 scratch

    const int n_edges = in_sizes[1] / 2;
    const int num_he  = out_size / HIDDEN;

    // 1) zero sums + counts
    int init_n = out_size > num_he ? out_size : num_he;
    ua_init_kernel<<<(init_n + 255) / 256, 256, 0, stream>>>(out, counts,
                                                             out_size, num_he);

    // 2) gather + scatter-add (one wave per edge, float4 per lane)
    long long total = (long long)n_edges * 32;
    int blocks = (int)((total + 255) / 256);
    ua_scatter_kernel<<<blocks, 256, 0, stream>>>(feats, hidx, out, counts,
                                                  n_edges);

    // 3) WMMA diagonal-scale epilogue: one wave per 16x16 tile
    int row_tiles = (num_he + 15) / 16;
    int waves = row_tiles * (HIDDEN / 16);
    long long threads = (long long)waves * 32;
    ua_finalize_wmma_kernel<<<(int)((threads + 255) / 256), 256, 0, stream>>>(
        out, counts, num_he);
}